// VIF_SSIM_Loss_65429531787463
// MI455X (gfx1250) — compile-verified
//
#include <hip/hip_runtime.h>

#define KS 11
#define WIDTH 1408
#define HEIGHT 1408
#define BATCH 16
#define NBLK_X 128                    // 1408 / 11
#define NBLK_Y 128
#define NBLOCKS_TOTAL (BATCH * NBLK_X * NBLK_Y)   // 262144
#define NPARTIAL (NBLK_Y * BATCH)                 // 2048 workgroup partials
#define C_CONST 0.0009f

typedef __attribute__((ext_vector_type(2))) float v2f;
typedef __attribute__((ext_vector_type(8))) float v8f;

// Cross-lane block reduction using the CDNA5 f32 WMMA path.
// partials is the B operand (4x16 f32): lane l<16 holds rows K=0 (.x) / K=1 (.y)
// for block column N=l; lanes 16..31 hold K=2 / K=3 for N=l-16.
// With A = ones(16x4):  D[m][n] = sum_k B[k][n]  -> every lane's d[0] is the
// complete 4-partial sum for block (lane & 15), broadcast across M.
__device__ __forceinline__ float block_sum_wmma(v2f partials) {
  v2f ones;
  ones.x = 1.0f;
  ones.y = 1.0f;
  v8f c = {};
  // 8 args: (neg_a, A, neg_b, B, c_mod, C, reuse_a, reuse_b)
  v8f d = __builtin_amdgcn_wmma_f32_16x16x4_f32(
      /*neg_a=*/false, ones, /*neg_b=*/false, partials,
      /*c_mod=*/(short)0, c, /*reuse_a=*/false, /*reuse_b=*/false);
  return d[0];
}

__global__ __launch_bounds__(256) void vif_ssim_stage1(
    const float* __restrict__ vis, const float* __restrict__ ir,
    const float* __restrict__ fus, float* __restrict__ partial) {
  const int lane = threadIdx.x & 31;
  const int wave = threadIdx.x >> 5;   // 0..7 (wave32)
  const int hf   = lane >> 4;          // half-wave -> which K pair of B
  const int bl   = lane & 15;          // block column within this wave's group
  const int rowblk = blockIdx.x;       // 0..127 block-row
  const int batch  = blockIdx.y;       // 0..15

  const size_t strip =
      (size_t)batch * (HEIGHT * WIDTH) + (size_t)rowblk * KS * WIDTH;
  const size_t base = strip + (size_t)((wave * 16 + bl) * KS);

  // 8 statistics x 2 per-lane partials (K partials of the WMMA B operand)
  v2f sv = {0.f, 0.f}, sv2 = {0.f, 0.f};
  v2f si = {0.f, 0.f}, si2 = {0.f, 0.f};
  v2f sf = {0.f, 0.f}, sf2 = {0.f, 0.f};
  v2f svf = {0.f, 0.f}, sif = {0.f, 0.f};

  // Pixel mapping: half-wave hf handles p % 4 in {2hf, 2hf+1}. Main loop does
  // 30 unconditional iterations (p <= 119 for all lanes); pixel 120 is peeled.
  // (row,col) of p0 = 4j + 2hf tracked incrementally: step 4 < KS=11 means at
  // most one row-wrap per iteration -> branchless cmp+select, no divmod.
  int cc  = 2 * hf;                    // col of p0 within block
  int off = 2 * hf;                    // row*WIDTH + col of p0
#pragma unroll 6
  for (int j = 0; j < 30; ++j) {
    const size_t i0 = base + (size_t)off;
    // p1 = p0 + 1: next col, or wrap to next row when cc == 10
    const size_t i1 = base + (size_t)(off + ((cc == KS - 1) ? (WIDTH - (KS - 1)) : 1));

    const float v0 = __builtin_nontemporal_load(vis + i0);
    const float i0v = __builtin_nontemporal_load(ir + i0);
    const float f0 = __builtin_nontemporal_load(fus + i0);
    const float v1 = __builtin_nontemporal_load(vis + i1);
    const float i1v = __builtin_nontemporal_load(ir + i1);
    const float f1 = __builtin_nontemporal_load(fus + i1);

    sv.x += v0;   sv2.x = fmaf(v0, v0, sv2.x);
    si.x += i0v;  si2.x = fmaf(i0v, i0v, si2.x);
    sf.x += f0;   sf2.x = fmaf(f0, f0, sf2.x);
    svf.x = fmaf(v0, f0, svf.x);
    sif.x = fmaf(i0v, f0, sif.x);

    sv.y += v1;   sv2.y = fmaf(v1, v1, sv2.y);
    si.y += i1v;  si2.y = fmaf(i1v, i1v, si2.y);
    sf.y += f1;   sf2.y = fmaf(f1, f1, sf2.y);
    svf.y = fmaf(v1, f1, svf.y);
    sif.y = fmaf(i1v, f1, sif.y);

    const int cc4 = cc + 4;
    const bool wrap = cc4 >= KS;
    off += wrap ? (WIDTH + 4 - KS) : 4;
    cc = wrap ? (cc4 - KS) : cc4;
  }
  // Tail: pixel p = 120 (row 10, col 10) belongs to half-wave 0, partial .x
  if (hf == 0) {
    const size_t it = base + (size_t)(10 * WIDTH + 10);
    const float v = __builtin_nontemporal_load(vis + it);
    const float i = __builtin_nontemporal_load(ir + it);
    const float f = __builtin_nontemporal_load(fus + it);
    sv.x += v;  sv2.x = fmaf(v, v, sv2.x);
    si.x += i;  si2.x = fmaf(i, i, si2.x);
    sf.x += f;  sf2.x = fmaf(f, f, sf2.x);
    svf.x = fmaf(v, f, svf.x);
    sif.x = fmaf(i, f, sif.x);
  }

  // Cross-lane reduction: one v_wmma_f32_16x16x4_f32 per statistic.
  // Uniform control flow here -> EXEC all ones as WMMA requires.
  const float bs_v  = block_sum_wmma(sv);
  const float bs_v2 = block_sum_wmma(sv2);
  const float bs_i  = block_sum_wmma(si);
  const float bs_i2 = block_sum_wmma(si2);
  const float bs_f  = block_sum_wmma(sf);
  const float bs_f2 = block_sum_wmma(sf2);
  const float bs_vf = block_sum_wmma(svf);
  const float bs_if = block_sum_wmma(sif);

  // Per-block score; every lane holds the stats of block (lane & 15).
  // Branchless: select the (mean, cov, var) inputs first, divide once.
  const float inv_n = 1.0f / (float)(KS * KS);
  const float vm = bs_v * inv_n;
  const float im = bs_i * inv_n;
  const float fm = bs_f * inv_n;
  const float vv = fabsf(fmaf(-vm, vm, bs_v2 * inv_n));
  const float iv = fabsf(fmaf(-im, im, bs_i2 * inv_n));
  const float fv = fabsf(fmaf(-fm, fm, bs_f2 * inv_n));
  const float vf_cov = fmaf(-vm, fm, bs_vf * inv_n);
  const float if_cov = fmaf(-im, fm, bs_if * inv_n);
  const float c = C_CONST;

  const bool sel = vm > im;
  const float m   = sel ? vm : im;        // chosen source mean
  const float var = sel ? vv : iv;        // chosen source variance
  const float cov = sel ? vf_cov : if_cov;
  const float ssim_l = fmaf(2.0f * m, fm, c) / (fmaf(m, m, fmaf(fm, fm, c)));
  const float ssim_s = (cov + c) / (var + fv + c);
  const float score = ssim_l * ssim_s;

  // Workgroup reduction: each block counted once (half-wave 0 only).
  __shared__ float red[256];
  red[threadIdx.x] = (hf == 0) ? score : 0.0f;
  __syncthreads();
  for (int s = 128; s > 0; s >>= 1) {
    if ((int)threadIdx.x < s) red[threadIdx.x] += red[threadIdx.x + s];
    __syncthreads();
  }
  if (threadIdx.x == 0)
    partial[blockIdx.y * gridDim.x + blockIdx.x] = red[0];
}

__global__ __launch_bounds__(256) void vif_ssim_stage2(
    const float* __restrict__ partial, float* __restrict__ out) {
  __shared__ float red[256];
  float s = 0.0f;
  for (int i = threadIdx.x; i < NPARTIAL; i += 256) s += partial[i];
  red[threadIdx.x] = s;
  __syncthreads();
  for (int st = 128; st > 0; st >>= 1) {
    if ((int)threadIdx.x < st) red[threadIdx.x] += red[threadIdx.x + st];
    __syncthreads();
  }
  if (threadIdx.x == 0)
    out[0] = 1.0f - red[0] / (float)NBLOCKS_TOTAL;
}

extern "C" void kernel_launch(void* const* d_in, const int* in_sizes, int n_in,
                              void* d_out, int out_size, void* d_ws,
                              size_t ws_size, hipStream_t stream) {
  const float* vis = (const float*)d_in[0];
  const float* ir  = (const float*)d_in[1];
  const float* fus = (const float*)d_in[2];
  float* partial = (float*)d_ws;  // 2048 floats = 8 KB scratch
  float* out = (float*)d_out;

  dim3 grid(NBLK_Y, BATCH);  // 128 block-rows x 16 batch images = 2048 WGs
  vif_ssim_stage1<<<grid, 256, 0, stream>>>(vis, ir, fus, partial);
  vif_ssim_stage2<<<1, 256, 0, stream>>>(partial, out);
}